// GaussianRenderer_7791070675876
// MI455X (gfx1250) — compile-verified
//
#include <hip/hip_runtime.h>
#include <math.h>

typedef __attribute__((ext_vector_type(2))) float v2f;
typedef __attribute__((ext_vector_type(8))) float v8f;

#define NG   512
#define IMW  256
#define IMH  256
#define NPIX (IMW * IMH)
#define NCHUNK (NG / 4)

// ---- workspace layout (float offsets) ----
#define U_KEY 0
#define U_MX   (1 * NG)
#define U_MY   (2 * NG)
#define U_I00  (3 * NG)
#define U_I01  (4 * NG)
#define U_I11  (5 * NG)
#define U_OP   (6 * NG)
#define U_XLO  (7 * NG)
#define U_XHI  (8 * NG)
#define U_YLO  (9 * NG)
#define U_YHI  (10 * NG)
#define U_VIS  (11 * NG)
#define U_CR   (12 * NG)
#define U_CG   (13 * NG)
#define U_CB   (14 * NG)
#define U_DEP  (15 * NG)
#define S_GEO  (16 * NG)            // 512 * 12 floats (3x float4 per gaussian, sorted)
#define S_BT   (S_GEO + 12 * NG)    // 128 chunks * 32 lanes * 2 floats = 8192 (WMMA B operand)
// total = 28 * NG + ... = 22528 floats = 88 KB

// ---------------- per-gaussian preprocessing ----------------
__global__ void gs_prep(const float* __restrict__ xyz, const float* __restrict__ opacity,
                        const float* __restrict__ feat, const float* __restrict__ cov,
                        const float* __restrict__ Km, const float* __restrict__ w2c,
                        float* __restrict__ out_radii, float* __restrict__ ws)
{
    int i = blockIdx.x * blockDim.x + threadIdx.x;
    if (i >= NG) return;

    // zero-init the WMMA B table (columns 4..15 must stay zero; sort fills 0..3)
    #pragma unroll
    for (int z = 0; z < 16; ++z) ws[S_BT + i * 16 + z] = 0.0f;

    float R00 = w2c[0], R01 = w2c[1], R02 = w2c[2],  t0 = w2c[3];
    float R10 = w2c[4], R11 = w2c[5], R12 = w2c[6],  t1 = w2c[7];
    float R20 = w2c[8], R21 = w2c[9], R22 = w2c[10], t2 = w2c[11];

    float X = xyz[3 * i], Y = xyz[3 * i + 1], Z = xyz[3 * i + 2];
    float cx = R00 * X + R01 * Y + R02 * Z + t0;
    float cy = R10 * X + R11 * Y + R12 * Z + t1;
    float cz = R20 * X + R21 * Y + R22 * Z + t2;   // depth

    float fx = Km[0], cpx = Km[2], fy = Km[4], cpy = Km[5];
    float hz = fmaxf(cz, 1e-8f);
    float iz = 1.0f / hz;
    float mx = (fx * cx + cpx * cz) * iz;
    float my = (fy * cy + cpy * cz) * iz;

    // camera position = -R^T t  (rigid inverse of w2c)
    float px = -(R00 * t0 + R10 * t1 + R20 * t2);
    float py = -(R01 * t0 + R11 * t1 + R21 * t2);
    float pz = -(R02 * t0 + R12 * t1 + R22 * t2);
    float vx = X - px, vy = Y - py, vz = Z - pz;
    float nrm = fmaxf(sqrtf(vx * vx + vy * vy + vz * vz), 1e-12f);
    vx /= nrm; vy /= nrm; vz /= nrm;

    // ---- SH (degree 3, 16 coeffs) -> RGB ----
    float xx = vx * vx, yy = vy * vy, zz = vz * vz;
    float xy_ = vx * vy, yz_ = vy * vz, xz_ = vx * vz;
    float rgb[3];
    #pragma unroll
    for (int c = 0; c < 3; ++c) {
        const float* f = feat + (size_t)i * 48 + c;
        float s0 = f[0], s1 = f[3], s2 = f[6], s3 = f[9];
        float s4 = f[12], s5 = f[15], s6 = f[18], s7 = f[21], s8 = f[24];
        float s9 = f[27], s10 = f[30], s11 = f[33], s12 = f[36], s13 = f[39], s14 = f[42], s15 = f[45];
        float r = 0.28209479177387814f * s0
                + 0.4886025119029199f * (-vy * s1 + vz * s2 - vx * s3)
                + 1.0925484305920792f * xy_ * s4
                - 1.0925484305920792f * yz_ * s5
                + 0.31539156525252005f * (2.0f * zz - xx - yy) * s6
                - 1.0925484305920792f * xz_ * s7
                + 0.5462742152960396f * (xx - yy) * s8
                - 0.5900435899266435f * vy * (3.0f * xx - yy) * s9
                + 2.890611442640554f * xy_ * vz * s10
                - 0.4570457994644658f * vy * (4.0f * zz - xx - yy) * s11
                + 0.3731763325901154f * vz * (2.0f * zz - 3.0f * xx - 3.0f * yy) * s12
                - 0.4570457994644658f * vx * (4.0f * zz - xx - yy) * s13
                + 1.445305721320277f * vz * (xx - yy) * s14
                - 0.5900435899266435f * vx * (xx - 3.0f * yy) * s15;
        rgb[c] = fminf(fmaxf(r + 0.5f, 0.0f), 1.0f);
    }

    // ---- 2D covariance: cov2d = (J R) cov (J R)^T + 0.3 I ----
    float tz = hz, tz2 = tz * tz;
    float J00 = fx / tz, J02 = -fx * cx / tz2;
    float J11 = fy / tz, J12 = -fy * cy / tz2;
    float T0x = J00 * R00 + J02 * R20, T0y = J00 * R01 + J02 * R21, T0z = J00 * R02 + J02 * R22;
    float T1x = J11 * R10 + J12 * R20, T1y = J11 * R11 + J12 * R21, T1z = J11 * R12 + J12 * R22;
    const float* M = cov + (size_t)i * 9;
    float v0x = M[0] * T0x + M[1] * T0y + M[2] * T0z;
    float v0y = M[3] * T0x + M[4] * T0y + M[5] * T0z;
    float v0z = M[6] * T0x + M[7] * T0y + M[8] * T0z;
    float v1x = M[0] * T1x + M[1] * T1y + M[2] * T1z;
    float v1y = M[3] * T1x + M[4] * T1y + M[5] * T1z;
    float v1z = M[6] * T1x + M[7] * T1y + M[8] * T1z;
    float a = T0x * v0x + T0y * v0y + T0z * v0z + 0.3f;
    float b = T0x * v1x + T0y * v1y + T0z * v1z;
    float d = T1x * v1x + T1y * v1y + T1z * v1z + 0.3f;

    bool visible = (cz > 0.01f) && (mx > -256.0f) && (mx < (float)IMW + 256.0f)
                                && (my > -256.0f) && (my < (float)IMH + 256.0f);

    float det = a * d - b * b;
    float trace = a + d;
    float lam = 0.5f * (trace + sqrtf(fmaxf(trace * trace - 4.0f * det, 0.0f)));
    float radf = 3.0f * sqrtf(fmaxf(lam, 1e-8f));
    out_radii[i] = visible ? radf : 0.0f;

    float detc = fmaxf(det, 1e-10f);
    float i00 = d / detc, i01 = -b / detc, i11 = a / detc;
    float radc = fmaxf(radf, 1.0f);
    float xlo = floorf(fminf(fmaxf(mx - radc, 0.0f), (float)IMW));
    float xhi = floorf(fminf(fmaxf(mx + radc + 1.0f, 0.0f), (float)IMW));
    float ylo = floorf(fminf(fmaxf(my - radc, 0.0f), (float)IMH));
    float yhi = floorf(fminf(fmaxf(my + radc + 1.0f, 0.0f), (float)IMH));

    ws[U_KEY + i] = visible ? cz : __builtin_inff();
    ws[U_MX + i] = mx;   ws[U_MY + i] = my;
    ws[U_I00 + i] = i00; ws[U_I01 + i] = i01; ws[U_I11 + i] = i11;
    ws[U_OP + i] = opacity[i];
    ws[U_XLO + i] = xlo; ws[U_XHI + i] = xhi;
    ws[U_YLO + i] = ylo; ws[U_YHI + i] = yhi;
    ws[U_VIS + i] = visible ? 1.0f : 0.0f;
    ws[U_CR + i] = rgb[0]; ws[U_CG + i] = rgb[1]; ws[U_CB + i] = rgb[2];
    ws[U_DEP + i] = cz;
}

// ------- stable depth rank-sort + scatter to packed geometry and WMMA B table -------
__global__ void gs_sort(float* __restrict__ ws)
{
    int i = blockIdx.x * blockDim.x + threadIdx.x;
    if (i >= NG) return;
    float ki = ws[U_KEY + i];
    int r = 0;
    for (int j = 0; j < NG; ++j) {
        float kj = ws[U_KEY + j];
        r += (kj < ki) || (kj == ki && j < i);
    }
    float vis = ws[U_VIS + i];
    float op  = (vis != 0.0f) ? ws[U_OP + i] : 0.0f;   // fold visibility into opacity
    float* g = ws + S_GEO + r * 12;
    g[0]  = ws[U_MX + i];  g[1]  = ws[U_MY + i];
    g[2]  = ws[U_I00 + i]; g[3]  = ws[U_I01 + i];
    g[4]  = ws[U_I11 + i]; g[5]  = op;
    g[6]  = ws[U_XLO + i]; g[7]  = ws[U_XHI + i];
    g[8]  = ws[U_YLO + i]; g[9]  = ws[U_YHI + i];
    g[10] = 0.0f;          g[11] = 0.0f;

    // WMMA B operand: chunk = r/4, row k = r%4 -> (lane half, slot), columns m=0..3 = r,g,b,depth
    float chan[4];
    chan[0] = ws[U_CR + i]; chan[1] = ws[U_CG + i];
    chan[2] = ws[U_CB + i]; chan[3] = ws[U_DEP + i];
    int chunk = r >> 2, k = r & 3;
    int half = (k >> 1) * 16, slot = k & 1;
    float* bt = ws + S_BT + chunk * 64;
    #pragma unroll
    for (int m = 0; m < 4; ++m) bt[(half + m) * 2 + slot] = chan[m];
}

// ---------------- WMMA compositor: one wave32 per 16-pixel tile ----------------
__global__ void __launch_bounds__(256) gs_render(const float* __restrict__ ws,
                                                 const float* __restrict__ bg,
                                                 float* __restrict__ out)
{
    int lane = threadIdx.x & 31;
    int wave = threadIdx.x >> 5;
    int tile = blockIdx.x * 8 + wave;        // 512 blocks * 8 waves = 4096 tiles
    int m    = lane & 15;
    int p    = tile * 16 + m;
    float fxp = (float)(p & (IMW - 1));
    float fyp = (float)(p >> 8);
    bool  hi  = lane >= 16;

    const float4* geo = (const float4*)(ws + S_GEO);
    const v2f*    bt  = (const v2f*)(ws + S_BT);

    v8f acc = {};          // D: 16 pixels x {r,g,b,depth,0...}
    float T = 1.0f;        // transmittance for pixel m (duplicated in both lane halves)

    #pragma unroll 4
    for (int chunk = 0; chunk < NCHUNK; ++chunk) {
        int gb = chunk * 4;
        __builtin_prefetch(&geo[(gb + 4) * 3], 0, 1);     // global_prefetch next chunk
        v2f bv = bt[chunk * 32 + lane];                    // one coalesced b64 load
        float w0, w1, w2, w3;
        #pragma unroll
        for (int k = 0; k < 4; ++k) {
            int g = gb + k;
            float4 g0 = geo[g * 3 + 0];      // mx, my, i00, i01
            float4 g1 = geo[g * 3 + 1];      // i11, op, xlo, xhi
            float4 g2 = geo[g * 3 + 2];      // ylo, yhi, -, -
            float dx = fxp - g0.x, dy = fyp - g0.y;
            float maha = g0.z * dx * dx + 2.0f * g0.w * dx * dy + g1.x * dy * dy;
            float wg = __expf(-0.5f * maha);
            float alpha = fminf(g1.y * wg, 0.99f);
            bool inb = (fxp >= g1.z) && (fxp < g1.w) && (fyp >= g2.x) && (fyp < g2.y);
            alpha = inb ? alpha : 0.0f;
            float wk = T * alpha;
            T *= (1.0f - alpha);
            if (k == 0) w0 = wk; else if (k == 1) w1 = wk; else if (k == 2) w2 = wk; else w3 = wk;
        }
        // A (16x4 f32): lanes 0-15 hold K=0,1 ; lanes 16-31 hold K=2,3 for row M=lane&15
        v2f av; av.x = hi ? w2 : w0; av.y = hi ? w3 : w1;
        acc = __builtin_amdgcn_wmma_f32_16x16x4_f32(
            /*neg_a=*/false, av, /*neg_b=*/false, bv,
            /*c_mod=*/(short)0, acc, /*reuse_a=*/false, /*reuse_b=*/false);
    }

    // broadcast per-pixel final transmittance before any divergence (EXEC all-1s)
    int mbase = hi ? 8 : 0;
    float Tm[8];
    #pragma unroll
    for (int j = 0; j < 8; ++j) Tm[j] = __shfl(T, mbase + j, 32);

    // D layout: VGPR j holds row M = mbase + j, lane column n = lane&15
    if (m < 4) {
        #pragma unroll
        for (int j = 0; j < 8; ++j) {
            int mm = mbase + j;
            int pp = tile * 16 + mm;
            float v = acc[j];
            if (m < 3) out[m * NPIX + pp] = v + bg[m] * Tm[j];   // render channels
            else       out[3 * NPIX + pp] = v;                    // depth image
        }
    }
    if (!hi) out[4 * NPIX + p] = 1.0f - T;                        // alpha image
}

extern "C" void kernel_launch(void* const* d_in, const int* in_sizes, int n_in,
                              void* d_out, int out_size, void* d_ws, size_t ws_size,
                              hipStream_t stream)
{
    const float* xyz      = (const float*)d_in[0];
    const float* opacity  = (const float*)d_in[1];
    const float* feat     = (const float*)d_in[2];
    const float* cov      = (const float*)d_in[3];
    const float* Km       = (const float*)d_in[4];
    const float* w2c      = (const float*)d_in[5];
    const float* bg       = (const float*)d_in[6];
    float* out = (float*)d_out;
    float* ws  = (float*)d_ws;   // needs 88 KB

    gs_prep<<<2, 256, 0, stream>>>(xyz, opacity, feat, cov, Km, w2c,
                                   out + 5 * NPIX /* radii @ 327680 */, ws);
    gs_sort<<<2, 256, 0, stream>>>(ws);
    gs_render<<<512, 256, 0, stream>>>(ws, bg, out);
}